// MessagePN2_49649821942251
// MI455X (gfx1250) — compile-verified
//
#include <hip/hip_runtime.h>
#include <hip/hip_bf16.h>

// Problem constants (match reference)
#define BB  8
#define NN  2048
#define CC  64
#define HH  8
#define DKK 3
#define HD  24            // H*DK
#define HID 1152
#define TWOC 128

typedef __attribute__((ext_vector_type(2))) float v2f;
typedef __attribute__((ext_vector_type(4))) float v4f;
typedef __attribute__((ext_vector_type(8))) float v8f;

// ---------------------------------------------------------------------------
// Kernel 1: q,k,v projections  ([B*N,64] @ [64,24] x3) -> ws
// ---------------------------------------------------------------------------
__global__ void qkv_kernel(const float* __restrict__ data,
                           const float* __restrict__ Wq,
                           const float* __restrict__ Wk,
                           const float* __restrict__ Wv,
                           float* __restrict__ qkv)  // q | k | v, each B*N*HD
{
    int tid = blockIdx.x * blockDim.x + threadIdx.x;
    const int total = BB * NN * 3 * HD;
    if (tid >= total) return;
    int row   = tid / (3 * HD);
    int c     = tid % (3 * HD);
    int which = c / HD;
    int j     = c % HD;
    const float* W = (which == 0) ? Wq : (which == 1) ? Wk : Wv;
    const float* x = data + (size_t)row * CC;
    float acc = 0.f;
#pragma unroll 8
    for (int i = 0; i < CC; ++i)
        acc = fmaf(x[i], W[i * HD + j], acc);
    qkv[(size_t)which * (BB * NN * HD) + (size_t)row * HD + j] = acc;
}

// ---------------------------------------------------------------------------
// Kernel 2: attention. One wave32 per score row (b,h,n).
// Online softmax. The attn matrix (1.07 GB) is the bandwidth-dominant output:
// each lane produces 4 consecutive probabilities and streams them out with a
// non-temporal global_store_b128 so the stream doesn't pollute L2 (which we
// want holding k/v and the MLP weights).
// ---------------------------------------------------------------------------
__global__ void attn_kernel(const float* __restrict__ q,
                            const float* __restrict__ k,
                            const float* __restrict__ v,
                            float* __restrict__ attn,  // [B,H,N,N]
                            float* __restrict__ o)     // [B,N,HD]
{
    const int lane = threadIdx.x & 31;
    const int wid  = threadIdx.x >> 5;
    const int row  = blockIdx.x * (blockDim.x >> 5) + wid;  // (b*H+h)*N + n
    const int n = row % NN;
    const int h = (row / NN) % HH;
    const int b = row / (NN * HH);

    const float scale = 0.57735026918962576451f;  // 1/sqrt(3)
    const float* qp = q + ((size_t)(b * NN + n)) * HD + h * DKK;
    const float q0 = qp[0] * scale, q1 = qp[1] * scale, q2 = qp[2] * scale;
    const float* kb = k + ((size_t)b * NN) * HD + h * DKK;
    const float* vb = v + ((size_t)b * NN) * HD + h * DKK;

    // pass 1: per-lane online max / sum, 4 scores per lane per iteration
    float lm = -3.4e38f, ls = 0.f;
    for (int m0 = 0; m0 < NN; m0 += 128) {
        const int m = m0 + lane * 4;
        float s[4];
#pragma unroll
        for (int j = 0; j < 4; ++j) {
            const float* kp = kb + (size_t)(m + j) * HD;
            s[j] = fmaf(q0, kp[0], fmaf(q1, kp[1], q2 * kp[2]));
        }
        float bm = fmaxf(fmaxf(s[0], s[1]), fmaxf(s[2], s[3]));
        float nm = fmaxf(lm, bm);
        ls = ls * __expf(lm - nm)
           + __expf(s[0] - nm) + __expf(s[1] - nm)
           + __expf(s[2] - nm) + __expf(s[3] - nm);
        lm = nm;
    }
    // wave32 combine of (max,sum) pairs
    for (int off = 16; off > 0; off >>= 1) {
        float om = __shfl_xor(lm, off, 32);
        float os = __shfl_xor(ls, off, 32);
        float nm = fmaxf(lm, om);
        ls = ls * __expf(lm - nm) + os * __expf(om - nm);
        lm = nm;
    }
    const float inv = 1.f / ls;

    // pass 2: recompute, normalize, NT-stream attn (b128), accumulate context
    float o0 = 0.f, o1 = 0.f, o2 = 0.f;
    float* ap = attn + (size_t)row * NN;
    for (int m0 = 0; m0 < NN; m0 += 128) {
        const int m = m0 + lane * 4;
        v4f pv;
#pragma unroll
        for (int j = 0; j < 4; ++j) {
            const float* kp = kb + (size_t)(m + j) * HD;
            float s = fmaf(q0, kp[0], fmaf(q1, kp[1], q2 * kp[2]));
            float p = __expf(s - lm) * inv;
            pv[j] = p;
            const float* vp = vb + (size_t)(m + j) * HD;
            o0 = fmaf(p, vp[0], o0);
            o1 = fmaf(p, vp[1], o1);
            o2 = fmaf(p, vp[2], o2);
        }
        __builtin_nontemporal_store(pv, (v4f*)(ap + m));  // global_store_b128 NT
    }
    for (int off = 16; off > 0; off >>= 1) {
        o0 += __shfl_xor(o0, off, 32);
        o1 += __shfl_xor(o1, off, 32);
        o2 += __shfl_xor(o2, off, 32);
    }
    if (lane == 0) {
        float* op = o + ((size_t)(b * NN + n)) * HD + h * DKK;
        op[0] = o0; op[1] = o1; op[2] = o2;
    }
}

// ---------------------------------------------------------------------------
// Kernel 3: fused MLP with V_WMMA_F32_16X16X4_F32.
// Per 16-row tile: x=[o@Wo | data] (16x128 in LDS) ->
//   h = relu(x@W1+b1) in 3 chunks of 384 cols (LDS) ->
//   y = relu(h@W2+b2) accumulated in registers across chunks.
// 4 waves / 128 threads per block. LDS ~33.5KB static.
// ---------------------------------------------------------------------------
#define XS 132                 // 128 + pad
#define CHUNK_T 24             // hidden tiles per chunk (24*16 = 384 cols)
#define CHUNK_N 384
#define HS 392                 // 384 + pad

__global__ void __launch_bounds__(128)
mlp_kernel(const float* __restrict__ data,
           const float* __restrict__ o,    // [B*N, HD]
           const float* __restrict__ Wo,   // [HD, C]
           const float* __restrict__ W1,   // [2C, HID]
           const float* __restrict__ b1,   // [HID]
           const float* __restrict__ W2,   // [HID, C]
           const float* __restrict__ b2,   // [C]
           float* __restrict__ y)          // [B*N, C]
{
    __shared__ float xs[16 * XS];
    __shared__ float hs[16 * HS];

    const int r0   = blockIdx.x * 16;
    const int tid  = threadIdx.x;
    const int lane = tid & 31;
    const int wid  = tid >> 5;     // 0..3
    const int m    = lane & 15;    // A row / B,D col within tile
    const int kh   = lane >> 4;    // K half select (0/1)

    // ---- Step A: build x tile = [o@Wo | data] in LDS ----
    for (int e = tid; e < 16 * CC; e += 128) {
        int r = e >> 6;
        int j = e & 63;
        const float* orow = o + (size_t)(r0 + r) * HD;
        float acc = 0.f;
#pragma unroll
        for (int t = 0; t < HD; ++t)
            acc = fmaf(orow[t], Wo[t * CC + j], acc);
        xs[r * XS + j]      = acc;                             // out part
        xs[r * XS + CC + j] = data[(size_t)(r0 + r) * CC + j]; // data part
    }
    __syncthreads();

    v8f yacc = {};                 // per-wave y tile accumulator (cols wid*16..)
    const int yncol = wid * 16 + m;

    for (int ch = 0; ch < HID / CHUNK_N; ++ch) {
        // ---- GEMM1 chunk: h[:, ch*384 .. +384) = relu(x @ W1 + b1) ----
        for (int t = wid; t < CHUNK_T; t += 4) {
            const int ct   = ch * CHUNK_T + t;
            const int ncol = ct * 16 + m;
            v8f acc = {};
#pragma unroll
            for (int kk = 0; kk < TWOC; kk += 4) {
                v2f a, bf;
                a.x  = xs[m * XS + kk + 2 * kh];
                a.y  = xs[m * XS + kk + 2 * kh + 1];
                bf.x = W1[(size_t)(kk + 2 * kh)     * HID + ncol];
                bf.y = W1[(size_t)(kk + 2 * kh + 1) * HID + ncol];
                if (kk + 8 < TWOC)
                    __builtin_prefetch(&W1[(size_t)(kk + 8) * HID + ncol], 0, 0);
                acc = __builtin_amdgcn_wmma_f32_16x16x4_f32(
                          false, a, false, bf, (short)0, acc, false, false);
            }
            float bias = b1[ncol];
#pragma unroll
            for (int r = 0; r < 8; ++r)
                hs[(r + 8 * kh) * HS + t * 16 + m] = fmaxf(acc[r] + bias, 0.f);
        }
        __syncthreads();

        // ---- GEMM2 partial: yacc += h_chunk @ W2[ch*384 .. , :] ----
        for (int kk = 0; kk < CHUNK_N; kk += 4) {
            v2f a, bf;
            a.x = hs[m * HS + kk + 2 * kh];
            a.y = hs[m * HS + kk + 2 * kh + 1];
            const int krow = ch * CHUNK_N + kk;
            bf.x = W2[(size_t)(krow + 2 * kh)     * CC + yncol];
            bf.y = W2[(size_t)(krow + 2 * kh + 1) * CC + yncol];
            yacc = __builtin_amdgcn_wmma_f32_16x16x4_f32(
                       false, a, false, bf, (short)0, yacc, false, false);
        }
        __syncthreads();   // hs reused next chunk
    }

    // ---- epilogue: bias + relu + store y tile ----
    float bias = b2[yncol];
#pragma unroll
    for (int r = 0; r < 8; ++r) {
        int row = r + 8 * kh;
        y[(size_t)(r0 + row) * CC + yncol] = fmaxf(yacc[r] + bias, 0.f);
    }
}

// ---------------------------------------------------------------------------
extern "C" void kernel_launch(void* const* d_in, const int* in_sizes, int n_in,
                              void* d_out, int out_size, void* d_ws, size_t ws_size,
                              hipStream_t stream) {
    (void)in_sizes; (void)n_in; (void)out_size; (void)ws_size;
    const float* data = (const float*)d_in[0];
    // d_in[1] = base_shape (scalar, unused)
    const float* Wq = (const float*)d_in[2];
    const float* Wk = (const float*)d_in[3];
    const float* Wv = (const float*)d_in[4];
    const float* Wo = (const float*)d_in[5];
    const float* W1 = (const float*)d_in[6];
    const float* b1 = (const float*)d_in[7];
    const float* W2 = (const float*)d_in[8];
    const float* b2 = (const float*)d_in[9];

    float* y    = (float*)d_out;                       // [B,N,C]
    float* attn = y + (size_t)BB * NN * CC;            // [B,H,N,N]

    float* ws = (float*)d_ws;
    float* q  = ws;
    float* k  = ws + 1 * (size_t)BB * NN * HD;
    float* v  = ws + 2 * (size_t)BB * NN * HD;
    float* o  = ws + 3 * (size_t)BB * NN * HD;

    {   // projections
        int total = BB * NN * 3 * HD;
        qkv_kernel<<<(total + 255) / 256, 256, 0, stream>>>(data, Wq, Wk, Wv, ws);
    }
    {   // attention: one wave per score row, 8 waves per block
        int rows = BB * HH * NN;
        attn_kernel<<<rows / 8, 256, 0, stream>>>(q, k, v, attn, o);
    }
    {   // fused MLP: 16-row tiles
        mlp_kernel<<<(BB * NN) / 16, 128, 0, stream>>>(data, o, Wo, W1, b1, W2, b2, y);
    }
}